// FoveationModule_4896262718103
// MI455X (gfx1250) — compile-verified
//
#include <hip/hip_runtime.h>
#include <cstdint>

typedef uint32_t u32;
typedef __attribute__((ext_vector_type(4))) u32   u32x4;
typedef __attribute__((ext_vector_type(8))) u32   u32x8;
typedef __attribute__((ext_vector_type(4))) float f32x4;

#define BATCH 128
#define CH    3
#define IMH   512
#define IMW   512
#define CROPD 128
#define OUTD  224

// One block per (batch, channel). TDM DMAs the 128x128 crop tile (row stride
// 512 floats) into LDS; 8 wave32s then bilinearly upsample it to 224x224,
// 4 pixels per thread per iteration, streamed out with non-temporal b128 stores.
__global__ __launch_bounds__(256) void foveation_kernel(
    const float* __restrict__ fixation,   // [B,2]
    const float* __restrict__ image,      // [B,C,H,W]
    float* __restrict__ out)              // [B,C,224,224]
{
    extern __shared__ float tile[];       // CROPD*CROPD floats = 64 KB
    const int bc = blockIdx.x;            // b*CH + c
    const int b  = bc / CH;

    // ---- Stage 1: wave 0 programs the Tensor Data Mover ----
    if (threadIdx.x < 32) {
        // Crop origin (reference swaps x/y: row from fixation[:,0], col from [:,1]).
        float fx = fixation[2 * b + 0];
        float fy = fixation[2 * b + 1];
        float left = fminf(fmaxf(fx * (float)IMH - 64.0f, 0.0f), (float)(IMH - CROPD));
        float top  = fminf(fmaxf(fy * (float)IMW - 64.0f, 0.0f), (float)(IMW - CROPD));
        int r0 = __builtin_amdgcn_readfirstlane((int)left);  // truncation == floor (nonneg)
        int c0 = __builtin_amdgcn_readfirstlane((int)top);

        uint64_t gaddr = (uint64_t)(uintptr_t)image +
                         4ull * ((uint64_t)bc * (IMH * IMW) +
                                 (uint64_t)r0 * IMW + (uint64_t)c0);
        u32 lds_base = (u32)(uintptr_t)&tile[0];   // generic->LDS: low 32 bits

        // D# group 0 (128b): count=1 | lds_addr | global_addr[56:0] | type=2
        u32x4 g0;
        g0.x = 1u;                                           // count=1, user desc
        g0.y = lds_base;                                     // lds_addr (bytes)
        g0.z = (u32)gaddr;                                   // global_addr[31:0]
        g0.w = ((u32)(gaddr >> 32) & 0x01FFFFFFu)            // global_addr[56:32]
               | (2u << 30);                                 // type = 2 ("image")

        // D# group 1 (256b): 2D tile descriptor.
        u32x8 g1;
        g1.s0 = (2u << 16);            // wg_mask=0 (not in cluster), data_size=4B
        g1.s1 = ((u32)IMW << 16);      // atomic_bar_addr=0 | tensor_dim0 lo16 = 512
        g1.s2 = ((u32)IMH << 16);      // tensor_dim0 hi = 0 | tensor_dim1 lo16 = 512
        g1.s3 = ((u32)CROPD << 16);    // tensor_dim1 hi = 0 | tile_dim0 = 128
        g1.s4 = (u32)CROPD;            // tile_dim1 = 128, tile_dim2 = 0 (2D)
        g1.s5 = (u32)IMW;              // tensor_dim0_stride[31:0] = 512
        g1.s6 = 0u;                    // stride0 hi16=0 | tensor_dim1_stride lo16=0
        g1.s7 = (u32)((u32)(IMH * IMW) >> 16); // tensor_dim1_stride[47:16] (unused, 2D)

        // 2D tensor: VADDR2/VADDR3 omitted (NULL).
        asm volatile("tensor_load_to_lds %0, %1" :: "s"(g0), "s"(g1) : "memory");
        __builtin_amdgcn_s_wait_tensorcnt(0);
    }
    __syncthreads();

    // ---- Stage 2: clamp-to-edge bilinear 128 -> 224 from LDS ----
    // 224*224 px = 12544 float4 = 49 iterations * 256 threads, no tail.
    const float sc = (float)CROPD / (float)OUTD;   // 4/7
    f32x4* __restrict__ out4 =
        (f32x4*)(out + (uint64_t)bc * (OUTD * OUTD));

    #pragma unroll 1
    for (int k = 0; k < 49; ++k) {
        int i4  = threadIdx.x + k * 256;       // float4 index within the tile
        int oy  = i4 / (OUTD / 4);             // /56
        int ox0 = (i4 - oy * (OUTD / 4)) * 4;  // first of 4 consecutive pixels

        // Row interpolation terms are shared by all 4 pixels.
        float sy = fminf(fmaxf(((float)oy + 0.5f) * sc - 0.5f, 0.0f), (float)(CROPD - 1));
        int   y0 = (int)sy;
        float wy = sy - (float)y0;
        int   y1 = min(y0 + 1, CROPD - 1);
        const float* rowA = &tile[y0 * CROPD];
        const float* rowB = &tile[y1 * CROPD];
        float cy0 = 1.0f - wy;

        f32x4 r;
        #pragma unroll
        for (int j = 0; j < 4; ++j) {
            float sx = fminf(fmaxf(((float)(ox0 + j) + 0.5f) * sc - 0.5f, 0.0f),
                             (float)(CROPD - 1));
            int   x0 = (int)sx;
            float wx = sx - (float)x0;
            int   x1 = min(x0 + 1, CROPD - 1);
            float cx0 = 1.0f - wx;
            float t = rowA[x0] * cx0 + rowA[x1] * wx;
            float u = rowB[x0] * cx0 + rowB[x1] * wx;
            r[j] = t * cy0 + u * wy;
        }
        // Output is write-once / never re-read: stream it past L2.
        __builtin_nontemporal_store(r, &out4[i4]);
    }
}

extern "C" void kernel_launch(void* const* d_in, const int* in_sizes, int n_in,
                              void* d_out, int out_size, void* d_ws, size_t ws_size,
                              hipStream_t stream) {
    const float* fixation = (const float*)d_in[0];   // [128,2]
    const float* image    = (const float*)d_in[1];   // [128,3,512,512]
    float* out            = (float*)d_out;           // [128,3,224,224]

    dim3 grid(BATCH * CH);   // 384 blocks, one crop tile each
    dim3 block(256);         // 8 wave32s
    size_t lds_bytes = (size_t)CROPD * CROPD * sizeof(float);  // 64 KB
    foveation_kernel<<<grid, block, lds_bytes, stream>>>(fixation, image, out);
}